// LinkPredModel_47699906789907
// MI455X (gfx1250) — compile-verified
//
#include <hip/hip_runtime.h>

#define N_NODES 50000
#define N_EDGES 800000
#define IN_F    128
#define HID_F   128
#define OUT_F   64

typedef __attribute__((ext_vector_type(2))) float v2f;
typedef __attribute__((ext_vector_type(8))) float v8f;

// ---------------------------------------------------------------- utilities
__global__ void zero_kernel(float* __restrict__ p, int n) {
    int i = blockIdx.x * blockDim.x + threadIdx.x;
    int stride = gridDim.x * blockDim.x;
    for (; i < n; i += stride) p[i] = 0.0f;
}

__global__ void degree_kernel(const int* __restrict__ dst,
                              float* __restrict__ deg, int E) {
    int e = blockIdx.x * blockDim.x + threadIdx.x;
    if (e < E) atomicAdd(&deg[dst[e]], 1.0f);
}

__global__ void invdeg_kernel(float* __restrict__ deg, int n) {
    int i = blockIdx.x * blockDim.x + threadIdx.x;
    if (i < n) deg[i] = 1.0f / fmaxf(deg[i], 1.0f);
}

// one wave32 per edge; each lane moves 4 consecutive floats (F = 128)
__global__ void scatter_kernel(const float* __restrict__ feat,
                               const int* __restrict__ src,
                               const int* __restrict__ dst,
                               float* __restrict__ agg, int E) {
    int gid  = blockIdx.x * blockDim.x + threadIdx.x;
    int e    = gid >> 5;
    int lane = gid & 31;
    if (e >= E) return;
    int s = src[e], d = dst[e];
    const float4 v = *(const float4*)(feat + (size_t)s * 128 + lane * 4);
    float* o = agg + (size_t)d * 128 + lane * 4;
    atomicAdd(o + 0, v.x);
    atomicAdd(o + 1, v.y);
    atomicAdd(o + 2, v.z);
    atomicAdd(o + 3, v.w);
}

// ------------------------------------------------------------- fused GEMM
// Out[16 rows x NCOLS] = act( Hself@Wself + (Aneigh*invdeg)@Wneigh + bias )
// One block per 16-row tile (M = 50000 = 3125*16, no guards -> EXEC all 1s).
// NCOLS/16 waves per block; each wave owns one 16x16 output tile and runs
// K=128 in steps of 4 with V_WMMA_F32_16X16X4_F32 (two WMMAs per step:
// self-term and neighbor-term share the f32 accumulator).
template <int NCOLS, bool RELU>
__global__ void sage_gemm_kernel(const float* __restrict__ Hself,   // [M,128]
                                 const float* __restrict__ Aneigh,  // [M,128]
                                 const float* __restrict__ invdeg,  // [M]
                                 const float* __restrict__ Wself,   // [128,NCOLS]
                                 const float* __restrict__ Wneigh,  // [128,NCOLS]
                                 const float* __restrict__ bias,    // [NCOLS]
                                 float* __restrict__ Out)           // [M,NCOLS]
{
    __shared__ float lds_s[16 * 128];
    __shared__ float lds_n[16 * 128];

    const int    tid     = threadIdx.x;
    const size_t rowbase = (size_t)blockIdx.x * 16;

    // stage the 16x128 A-tiles (coalesced 512B rows); fold mean scaling in
    for (int i = tid; i < 16 * 128; i += blockDim.x) {
        int r = i >> 7;
        size_t g = (rowbase + r) * 128 + (i & 127);
        lds_s[i] = Hself[g];
        lds_n[i] = Aneigh[g] * invdeg[rowbase + r];
    }
    __syncthreads();

    const int wave = tid >> 5;        // 16-column tile index
    const int lane = tid & 31;
    const int half = lane >> 4;       // 0 -> K pair {k0,k0+1}, 1 -> {k0+2,k0+3}
    const int l    = lane & 15;       // row (A) / col (B) within tile
    const int col  = wave * 16 + l;

    v8f acc = {};
#pragma unroll
    for (int k0 = 0; k0 < 128; k0 += 4) {
        const int ka = k0 + 2 * half;
        // A fragments: 32-bit A 16x4 layout -> ds_load_b64 per lane
        v2f a_s = *(const v2f*)&lds_s[l * 128 + ka];
        v2f a_n = *(const v2f*)&lds_n[l * 128 + ka];
        // B fragments: K rows k a / ka+1 striped across lanes (L2-hot weights)
        v2f b_s, b_n;
        b_s.x = Wself [(size_t)ka * NCOLS + col];
        b_s.y = Wself [(size_t)(ka + 1) * NCOLS + col];
        b_n.x = Wneigh[(size_t)ka * NCOLS + col];
        b_n.y = Wneigh[(size_t)(ka + 1) * NCOLS + col];
        acc = __builtin_amdgcn_wmma_f32_16x16x4_f32(
                  false, a_s, false, b_s, (short)0, acc, false, false);
        acc = __builtin_amdgcn_wmma_f32_16x16x4_f32(
                  false, a_n, false, b_n, (short)0, acc, false, false);
    }

    // epilogue: bias (+ ReLU), C/D layout: VGPR r -> row r (lanes 0-15) / r+8
    const float bv = bias[col];
#pragma unroll
    for (int r = 0; r < 8; ++r) {
        const int row = r + 8 * half;
        float v = acc[r] + bv;
        if (RELU) v = fmaxf(v, 0.0f);
        Out[(rowbase + row) * NCOLS + col] = v;
    }
}

// ------------------------------------------------- edge score: dot(h2[s],h2[d])
// one wave32 per edge; lane covers 2 of the 64 features; wave32 xor-reduce
__global__ void score_kernel(const float* __restrict__ h2,
                             const int* __restrict__ src,
                             const int* __restrict__ dst,
                             float* __restrict__ out, int E) {
    int gid  = blockIdx.x * blockDim.x + threadIdx.x;
    int e    = gid >> 5;
    int lane = gid & 31;
    if (e >= E) return;
    const float2 a = *(const float2*)(h2 + (size_t)src[e] * 64 + lane * 2);
    const float2 b = *(const float2*)(h2 + (size_t)dst[e] * 64 + lane * 2);
    float s = a.x * b.x + a.y * b.y;
#pragma unroll
    for (int off = 16; off > 0; off >>= 1)
        s += __shfl_xor(s, off, 32);
    if (lane == 0) out[e] = s;
}

// ------------------------------------------------------------------- launch
extern "C" void kernel_launch(void* const* d_in, const int* in_sizes, int n_in,
                              void* d_out, int out_size, void* d_ws, size_t ws_size,
                              hipStream_t stream) {
    const float* x   = (const float*)d_in[0];
    const int*   src = (const int*)  d_in[1];
    const int*   dst = (const int*)  d_in[2];
    const float* W1s = (const float*)d_in[3];
    const float* W1n = (const float*)d_in[4];
    const float* b1  = (const float*)d_in[5];
    const float* W2s = (const float*)d_in[6];
    const float* W2n = (const float*)d_in[7];
    const float* b2  = (const float*)d_in[8];
    float*       out = (float*)d_out;

    // workspace carve-up (floats): invdeg | agg (reused) | h1 | h2  (~64 MB)
    float* ws     = (float*)d_ws;
    float* invdeg = ws;
    float* agg    = ws + 50048;                       // 16B-aligned offset
    float* h1     = agg + (size_t)N_NODES * HID_F;
    float* h2     = h1  + (size_t)N_NODES * HID_F;

    const int EDGE_WAVE_BLOCKS = (N_EDGES * 32 + 255) / 256;

    // degrees -> inv_deg
    zero_kernel<<<1024, 256, 0, stream>>>(invdeg, N_NODES);
    degree_kernel<<<(N_EDGES + 255) / 256, 256, 0, stream>>>(dst, invdeg, N_EDGES);
    invdeg_kernel<<<(N_NODES + 255) / 256, 256, 0, stream>>>(invdeg, N_NODES);

    // layer 1: aggregate x then fused dual-GEMM + bias + ReLU
    zero_kernel<<<4096, 256, 0, stream>>>(agg, N_NODES * IN_F);
    scatter_kernel<<<EDGE_WAVE_BLOCKS, 256, 0, stream>>>(x, src, dst, agg, N_EDGES);
    sage_gemm_kernel<HID_F, true><<<N_NODES / 16, 256, 0, stream>>>(
        x, agg, invdeg, W1s, W1n, b1, h1);

    // layer 2: aggregate h1 then fused dual-GEMM + bias
    zero_kernel<<<4096, 256, 0, stream>>>(agg, N_NODES * HID_F);
    scatter_kernel<<<EDGE_WAVE_BLOCKS, 256, 0, stream>>>(h1, src, dst, agg, N_EDGES);
    sage_gemm_kernel<OUT_F, false><<<N_NODES / 16, 128, 0, stream>>>(
        h1, agg, invdeg, W2s, W2n, b2, h2);

    // edge-wise dot products
    score_kernel<<<EDGE_WAVE_BLOCKS, 256, 0, stream>>>(h2, src, dst, out, N_EDGES);
}